// Attention_38276748542287
// MI455X (gfx1250) — compile-verified
//
#include <hip/hip_runtime.h>

// Problem constants (match reference): B=32, S=2048, ENC_DIM=DEC_DIM=2048
#define B_   32
#define S_   2048
#define E_   2048

// Fused flash-style path parameters
#define NSPLIT  16                // S-chunks per batch -> 32*16 = 512 workgroups
#define RPW     (S_ / NSPLIT)     // rows per workgroup = 128
#define TILE    16                // rows staged in LDS per iteration
#define NTILE   (RPW / TILE)      // 8 tiles per workgroup
#define LSTRIDE 2056              // padded LDS row stride (floats). 2056 % 64 == 8,
                                  // so row r+2 is shifted 16 banks vs row r ->
                                  // lanes 0..15 (rows r,r+1) and lanes 16..31
                                  // (rows r+2,r+3) hit disjoint bank sets.

typedef __attribute__((ext_vector_type(2))) float v2f;
typedef __attribute__((ext_vector_type(4))) float v4f;
typedef __attribute__((ext_vector_type(8))) float v8f;

// Generic pointers to LDS carry the LDS byte offset in their low 32 bits
// that is what the async load's
// VDST (LDS-address VGPR) wants.
__device__ __forceinline__ unsigned lds_addr_of(const void* p) {
  return (unsigned)(unsigned long long)p;
}

// ===========================================================================
// FUSED PATH (single pass over enc_hs, ~512 MB HBM traffic, ~22 us roofline)
// ===========================================================================
//
// 512 threads = 16 waves per workgroup; workgroup owns (b, 128-row S-chunk).
// LDS holds TWO 16-row tile buffers (2 x 128.5 KB) + w_enc (8 KB) ~= 272 KB,
// so one WG fills a WGP — double buffering keeps ~128 KB of async DMA in
// flight per WGP while WMMA consumes the other buffer.
//
// Per 16-row tile:
//   1) issue async global->LDS DMA for the NEXT tile (ASYNCcnt), then
//      s_wait_asynccnt 16 so only the next tile's 16 ops/wave stay pending
//   2) wave-per-row dot with w_enc -> 16 scores (raw scores also written out)
//   3) online softmax: merge tile max into running (m,l), rescale accumulators
//   4) WMMA: D = A x B + C with V_WMMA_F32_16X16X4_F32, A[m,k]=exp(sc[k]-m)
//      broadcast over m, B[k,n]=lds_row[k][e0+n]; wave owns 128 e-columns
//      -> 8 v8f accumulators (64 persistent VGPRs, stays under the 256 window)
//
// The dec_h.w_dec + att_b scalar cancels exactly under softmax
// (softmax(x+c)==softmax(x)), so the fused path never touches dec_h.
// ---------------------------------------------------------------------------
__global__ __launch_bounds__(512) void fused_flash_kernel(
    const float* __restrict__ enc, const float* __restrict__ att_w,
    float* __restrict__ raw_scores, float* __restrict__ acc_p,
    float* __restrict__ m_p, float* __restrict__ l_p) {
  __shared__ float tile[2][TILE * LSTRIDE];  // 2 x ~128.5 KB
  __shared__ float w_enc[E_];                // 8 KB
  __shared__ float sc[TILE];

  const int wg  = blockIdx.x;
  const int b   = wg / NSPLIT;
  const int cs0 = (wg % NSPLIT) * RPW;

  const int tid  = threadIdx.x;
  const int wave = tid >> 5;     // 0..15
  const int lane = tid & 31;
  const int half = lane >> 4;    // 0: K pair {0,1}; 1: K pair {2,3}
  const int n    = lane & 15;
  const int koff = half * 2;

  for (int i = tid; i < E_; i += 512) w_enc[i] = att_w[E_ + i];

  // stage tile `tileidx` of this chunk into buffer `bufidx`
  // 8192 float4 chunks, 16 per thread (512 threads); GV addressing; ASYNCcnt.
#define ISSUE_TILE(bufidx, tileidx)                                          \
  do {                                                                       \
    const int s0i_ = cs0 + (tileidx) * TILE;                                 \
    _Pragma("unroll") for (int i_ = 0; i_ < 16; ++i_) {                      \
      const int idx_ = tid + i_ * 512;       /* [0, 8192) */                 \
      const int row_ = idx_ >> 9;            /* 512 float4 per row */        \
      const int col_ = (idx_ & 511) * 4;                                     \
      const unsigned la_ =                                                   \
          lds_addr_of(&tile[bufidx][row_ * LSTRIDE + col_]);                 \
      const float* ga_ =                                                     \
          enc + ((size_t)(b * S_ + s0i_ + row_)) * E_ + col_;                \
      asm volatile("global_load_async_to_lds_b128 %0, %1, off"               \
                   :: "v"(la_), "v"(ga_) : "memory");                        \
    }                                                                        \
  } while (0)

  v8f zero = {};
  v8f c[8];
#pragma unroll
  for (int t = 0; t < 8; ++t) c[t] = zero;
  float m = -3.402823466e+38f;
  float l = 0.0f;

  ISSUE_TILE(0, 0);

  for (int tt = 0; tt < NTILE; ++tt) {
    const int cur = tt & 1;
    const float* tl = &tile[cur][0];

    // ---- 1) prefetch next tile into the other buffer, drain current ----
    if (tt + 1 < NTILE) {
      ISSUE_TILE((tt + 1) & 1, tt + 1);
      asm volatile("s_wait_asynccnt 0x10" ::: "memory");  // current tile done
    } else {
      asm volatile("s_wait_asynccnt 0x0" ::: "memory");
    }
    __syncthreads();

    // ---- 2) scores: wave w handles staged row w ----
    {
      const int r = wave;
      float d = 0.0f;
#pragma unroll
      for (int i = 0; i < 16; ++i) {
        const int idx = lane * 4 + i * 128;
        v4f e = *(const v4f*)(&tl[r * LSTRIDE + idx]);
        v4f w = *(const v4f*)(&w_enc[idx]);
        d += e.x * w.x + e.y * w.y + e.z * w.z + e.w * w.w;
      }
#pragma unroll
      for (int off = 16; off >= 1; off >>= 1) d += __shfl_xor(d, off, 32);
      if (lane == 0) {
        sc[r] = d;
        raw_scores[(size_t)b * S_ + cs0 + tt * TILE + r] = d;
      }
    }
    __syncthreads();

    // ---- 3) online softmax update (every thread, redundantly) ----
    float tm = sc[0];
#pragma unroll
    for (int s = 1; s < TILE; ++s) tm = fmaxf(tm, sc[s]);
    const float newm  = fmaxf(m, tm);
    const float scale = __expf(m - newm);   // exp(-inf)=0 on first tile
    float ls = 0.0f;
#pragma unroll
    for (int s = 0; s < TILE; ++s) ls += __expf(sc[s] - newm);
    l = l * scale + ls;
    m = newm;
#pragma unroll
    for (int t = 0; t < 8; ++t) c[t] *= scale;

    // per-lane A-operand weights for the 4 K-groups of this tile
    v2f pa[4];
#pragma unroll
    for (int j = 0; j < 4; ++j) {
      pa[j].x = __expf(sc[4 * j + koff] - newm);
      pa[j].y = __expf(sc[4 * j + koff + 1] - newm);
    }

    // ---- 4) WMMA accumulate from LDS (wave owns cols wave*128..+127) ----
#pragma unroll
    for (int t = 0; t < 8; ++t) {
      const int e0 = wave * 128 + t * 16 + n;
#pragma unroll
      for (int j = 0; j < 4; ++j) {
        const int r = 4 * j + koff;
        v2f bb;
        bb.x = tl[r * LSTRIDE + e0];
        bb.y = tl[(r + 1) * LSTRIDE + e0];
        c[t] = __builtin_amdgcn_wmma_f32_16x16x4_f32(false, pa[j], false, bb,
                                                     (short)0, c[t], false, false);
      }
    }
    __syncthreads();   // all reads done before this buffer is re-filled
  }
#undef ISSUE_TILE

  // ---- write partials: row M=0 of D lives in element 0 of lanes 0..15 ----
  float* accw = acc_p + (size_t)wg * E_;
  if (lane < 16) {
#pragma unroll
    for (int t = 0; t < 8; ++t) accw[wave * 128 + t * 16 + n] = c[t][0];
  }
  if (tid == 0) { m_p[wg] = m; l_p[wg] = l; }
}

// attended[b,e] = sum_c acc_c[e]*exp(m_c-gm) / L ;  L = sum_c l_c*exp(m_c-gm)
__global__ __launch_bounds__(256) void combine_attended_kernel(
    const float* __restrict__ acc_p, const float* __restrict__ m_p,
    const float* __restrict__ l_p, float* __restrict__ attended) {
  const int b = blockIdx.x, tid = threadIdx.x;
  float mc[NSPLIT], fc[NSPLIT];
#pragma unroll
  for (int k = 0; k < NSPLIT; ++k) mc[k] = m_p[b * NSPLIT + k];
  float gm = mc[0];
#pragma unroll
  for (int k = 1; k < NSPLIT; ++k) gm = fmaxf(gm, mc[k]);
  float L = 0.0f;
#pragma unroll
  for (int k = 0; k < NSPLIT; ++k) {
    fc[k] = __expf(mc[k] - gm);
    L += l_p[b * NSPLIT + k] * fc[k];
  }
  const float inv = 1.0f / L;
#pragma unroll
  for (int i = 0; i < 8; ++i) {
    const int e = tid + i * 256;
    float s = 0.0f;
#pragma unroll
    for (int k = 0; k < NSPLIT; ++k)
      s += acc_p[((size_t)(b * NSPLIT + k)) * E_ + e] * fc[k];
    attended[(size_t)b * E_ + e] = s * inv;
  }
}

// att_weight[b,s] = exp(raw_score - gm) / L   (in place over raw scores)
__global__ __launch_bounds__(256) void finalize_weights_kernel(
    const float* __restrict__ m_p, const float* __restrict__ l_p,
    float* __restrict__ weights) {
  const int b = blockIdx.x, tid = threadIdx.x;
  float gm = m_p[b * NSPLIT];
#pragma unroll
  for (int k = 1; k < NSPLIT; ++k) gm = fmaxf(gm, m_p[b * NSPLIT + k]);
  float L = 0.0f;
#pragma unroll
  for (int k = 0; k < NSPLIT; ++k)
    L += l_p[b * NSPLIT + k] * __expf(m_p[b * NSPLIT + k] - gm);
  const float inv = 1.0f / L;
#pragma unroll
  for (int i = 0; i < 8; ++i) {
    const size_t idx = (size_t)b * S_ + tid + i * 256;
    weights[idx] = __expf(weights[idx] - gm) * inv;
  }
}

// ===========================================================================
// FALLBACK PATH (two passes over enc_hs; used if workspace is too small)
// ===========================================================================
__global__ __launch_bounds__(256) void scores_kernel(
    const float* __restrict__ enc, const float* __restrict__ att_w,
    float* __restrict__ scores) {
  __shared__ float w_enc[E_];
  for (int i = threadIdx.x; i < E_; i += 256) w_enc[i] = att_w[E_ + i];
  __syncthreads();

  const int wave = threadIdx.x >> 5;
  const int lane = threadIdx.x & 31;
  const long row = (long)blockIdx.x * 8 + wave;
  const float* rp = enc + row * (long)E_;

  float acc = 0.0f;
#pragma unroll
  for (int i = 0; i < 16; ++i) {
    const int idx = lane * 4 + i * 128;
    v4f e = *(const v4f*)(rp + idx);
    v4f w = *(const v4f*)(w_enc + idx);
    acc += e.x * w.x + e.y * w.y + e.z * w.z + e.w * w.w;
  }
#pragma unroll
  for (int off = 16; off >= 1; off >>= 1) acc += __shfl_xor(acc, off, 32);
  if (lane == 0) scores[row] = acc;
}

__global__ __launch_bounds__(256) void softmax_kernel(
    const float* __restrict__ dec_h, const float* __restrict__ att_w,
    const float* __restrict__ att_b, float* __restrict__ weights) {
  const int b   = blockIdx.x;
  const int tid = threadIdx.x;
  __shared__ float red[256];

  float p = 0.0f;
  for (int i = tid; i < E_; i += 256) p += dec_h[(long)b * E_ + i] * att_w[i];
  red[tid] = p;
  __syncthreads();
#pragma unroll
  for (int s = 128; s >= 1; s >>= 1) {
    if (tid < s) red[tid] += red[tid + s];
    __syncthreads();
  }
  const float base = red[0] + att_b[0];
  __syncthreads();

  float v[8];
  float m = -3.402823466e+38f;
#pragma unroll
  for (int i = 0; i < 8; ++i) {
    v[i] = weights[(long)b * S_ + tid + i * 256] + base;
    m = fmaxf(m, v[i]);
  }
  red[tid] = m;
  __syncthreads();
#pragma unroll
  for (int s = 128; s >= 1; s >>= 1) {
    if (tid < s) red[tid] = fmaxf(red[tid], red[tid + s]);
    __syncthreads();
  }
  m = red[0];
  __syncthreads();

  float lsum = 0.0f;
#pragma unroll
  for (int i = 0; i < 8; ++i) {
    v[i] = __expf(v[i] - m);
    lsum += v[i];
  }
  red[tid] = lsum;
  __syncthreads();
#pragma unroll
  for (int s = 128; s >= 1; s >>= 1) {
    if (tid < s) red[tid] += red[tid + s];
    __syncthreads();
  }
  const float inv = 1.0f / red[0];
#pragma unroll
  for (int i = 0; i < 8; ++i)
    weights[(long)b * S_ + tid + i * 256] = v[i] * inv;
}

__global__ __launch_bounds__(256) void attended_kernel(
    const float* __restrict__ enc, const float* __restrict__ weights,
    float* __restrict__ attended) {
  __shared__ float wrow[S_];
  const int b   = blockIdx.x >> 4;
  const int grp = blockIdx.x & 15;
  for (int i = threadIdx.x; i < S_; i += 256) wrow[i] = weights[(long)b * S_ + i];
  __syncthreads();

  const int wave = threadIdx.x >> 5;
  const int lane = threadIdx.x & 31;
  const int e0   = (grp * 8 + wave) * 16;
  const int half = lane >> 4;
  const int n    = lane & 15;

  const float* base = enc + (long)b * S_ * E_ + e0 + n;

  v8f c0 = {};
  v8f c1 = {};
  for (int s0 = 0; s0 < S_; s0 += 8) {
    const int k0 = s0 + half * 2;
    v2f a0, bm0;
    a0.x  = wrow[k0];
    a0.y  = wrow[k0 + 1];
    bm0.x = base[(long)k0 * E_];
    bm0.y = base[(long)(k0 + 1) * E_];
    c0 = __builtin_amdgcn_wmma_f32_16x16x4_f32(false, a0, false, bm0,
                                               (short)0, c0, false, false);
    const int k1 = s0 + 4 + half * 2;
    v2f a1, bm1;
    a1.x  = wrow[k1];
    a1.y  = wrow[k1 + 1];
    bm1.x = base[(long)k1 * E_];
    bm1.y = base[(long)(k1 + 1) * E_];
    c1 = __builtin_amdgcn_wmma_f32_16x16x4_f32(false, a1, false, bm1,
                                               (short)0, c1, false, false);
  }
  if (lane < 16) attended[(long)b * E_ + e0 + n] = c0[0] + c1[0];
}

// ===========================================================================
// Launch. d_out layout: [attended (B*E) | att_weight (B*S)].
// ===========================================================================
extern "C" void kernel_launch(void* const* d_in, const int* in_sizes, int n_in,
                              void* d_out, int out_size, void* d_ws,
                              size_t ws_size, hipStream_t stream) {
  (void)in_sizes; (void)n_in; (void)out_size;
  const float* dec_h = (const float*)d_in[0];
  const float* enc   = (const float*)d_in[1];
  const float* att_w = (const float*)d_in[2];
  const float* att_b = (const float*)d_in[3];

  float* attended = (float*)d_out;
  float* weights  = (float*)d_out + (size_t)B_ * E_;   // scores staged here

  const size_t need =
      ((size_t)NSPLIT * B_ * E_ + 2ull * NSPLIT * B_) * sizeof(float);

  if (d_ws != nullptr && ws_size >= need) {
    // Single pass over enc_hs (~512 MB HBM traffic, ~22 us roofline)
    float* acc_p = (float*)d_ws;
    float* m_p   = acc_p + (size_t)NSPLIT * B_ * E_;
    float* l_p   = m_p + NSPLIT * B_;
    fused_flash_kernel<<<B_ * NSPLIT, 512, 0, stream>>>(enc, att_w, weights,
                                                        acc_p, m_p, l_p);
    combine_attended_kernel<<<B_, 256, 0, stream>>>(acc_p, m_p, l_p, attended);
    finalize_weights_kernel<<<B_, 256, 0, stream>>>(m_p, l_p, weights);
  } else {
    // Two passes over enc_hs (~1 GB HBM traffic, ~44 us roofline)
    scores_kernel<<<(B_ * S_) / 8, 256, 0, stream>>>(enc, att_w, weights);
    softmax_kernel<<<B_, 256, 0, stream>>>(dec_h, att_w, att_b, weights);
    attended_kernel<<<B_ * 16, 256, 0, stream>>>(enc, weights, attended);
  }
}